// projection_6708738916627
// MI455X (gfx1250) — compile-verified
//
#include <hip/hip_runtime.h>

typedef __attribute__((ext_vector_type(2))) float v2f;
typedef __attribute__((ext_vector_type(8))) float v8f;

#define VD 128              // volume D
#define VW 128              // volume W
#define VH 128              // volume H (contiguous)
#define RES 180             // detector resolution (both dims)
#define KS 128              // samples per ray
#define JTILES 12           // ceil(180/16) ray tiles per detector row
#define NBATCH 4

// Trilinear sample with zero padding outside [0,127]^3.
// y is always in range (iy = k*127/128), only x (H dim) and z (D dim) checked.
__device__ __forceinline__ float tri_sample(const float* __restrict__ vol,
                                            float ixf, int y0, float wy0, float wy1,
                                            float izf) {
    float xf = floorf(ixf);
    float zf = floorf(izf);
    float wx1 = ixf - xf, wx0 = 1.0f - wx1;
    float wz1 = izf - zf, wz0 = 1.0f - wz1;
    int x0 = (int)xf, z0 = (int)zf;
    int x1 = x0 + 1,  z1 = z0 + 1;
    bool vx0 = (unsigned)x0 < 128u;
    bool vx1 = (unsigned)x1 < 128u;
    bool vz0 = (unsigned)z0 < 128u;
    bool vz1 = (unsigned)z1 < 128u;
    int xc0 = min(max(x0, 0), 127), xc1 = min(max(x1, 0), 127);
    int zc0 = min(max(z0, 0), 127), zc1 = min(max(z1, 0), 127);
    int b00 = (zc0 * VW + y0) * VH;      // z0, y0 row base
    int b10 = (zc1 * VW + y0) * VH;      // z1, y0 row base
    // Always-load (clamped, in-range) + select-to-zero: keeps code branch-free.
    float l000 = vol[b00 + xc0];
    float l001 = vol[b00 + xc1];
    float l010 = vol[b00 + VH + xc0];
    float l011 = vol[b00 + VH + xc1];
    float l100 = vol[b10 + xc0];
    float l101 = vol[b10 + xc1];
    float l110 = vol[b10 + VH + xc0];
    float l111 = vol[b10 + VH + xc1];
    float c000 = (vz0 && vx0) ? l000 : 0.0f;
    float c001 = (vz0 && vx1) ? l001 : 0.0f;
    float c010 = (vz0 && vx0) ? l010 : 0.0f;
    float c011 = (vz0 && vx1) ? l011 : 0.0f;
    float c100 = (vz1 && vx0) ? l100 : 0.0f;
    float c101 = (vz1 && vx1) ? l101 : 0.0f;
    float c110 = (vz1 && vx0) ? l110 : 0.0f;
    float c111 = (vz1 && vx1) ? l111 : 0.0f;
    float s00 = fmaf(c001, wx1, c000 * wx0);   // z0,y0
    float s01 = fmaf(c011, wx1, c010 * wx0);   // z0,y1
    float s10 = fmaf(c101, wx1, c100 * wx0);   // z1,y0
    float s11 = fmaf(c111, wx1, c110 * wx0);   // z1,y1
    float s0  = fmaf(s01, wy1, s00 * wy0);
    float s1  = fmaf(s11, wy1, s10 * wy0);
    return fmaf(s1, wz1, s0 * wz0);
}

// One wave = 16 rays (consecutive j) x full K march.
// K-reduction done via V_WMMA_F32_16X16X4_F32 with B = ones:
//   A[m][k] = dx_m * sample(ray m, k); D = A*1 + C accumulates row sums.
// A-fragment layout (32-bit A 16x4): lanes 0-15 hold K=0,1 in v[0],v[1];
// lanes 16-31 hold K=2,3. Each lane computes exactly its own fragment slots,
// so no cross-lane data movement is needed.
__global__ __launch_bounds__(256) void projection_kernel(
        const float* __restrict__ vol,
        const float* __restrict__ poses,
        const float* __restrict__ spacing,
        float* __restrict__ out) {
    const int tid  = threadIdx.x;
    const int lane = tid & 31;
    const int wave = blockIdx.x * 8 + (tid >> 5);

    const int jt  = wave % JTILES;
    const int rem = wave / JTILES;
    const int i   = rem % RES;
    const int b   = rem / RES;

    const int m  = lane & 15;            // ray-in-tile this lane samples for
    const int kb = (lane >> 4) << 1;     // local-K base: 0 (lanes<16) or 2

    const int j0 = jt * 16;
    const int j  = j0 + m;               // may exceed 179 on the last tile (dummy ray)

    const float ex = poses[b * 3 + 0];
    const float ey = poses[b * 3 + 1];
    const float ez = poses[b * 3 + 2];
    const float sp0 = spacing[0], sp1 = spacing[1], sp2 = spacing[2];

    const float Ix = (float)(i - 90) - ex;
    const float Iy = -ey;                                // detector plane y = 0
    const float Iz = (float)(j - 90) - ez;
    const float invIy = 1.0f / Iy;
    const float a0 = Ix * invIy, a2 = Iz * invIy;
    const float dx = sqrtf(a0 * a0 * sp0 * sp0 + sp1 * sp1 + a2 * a2 * sp2 * sp2);

    const float cyc = 127.0f / 128.0f;   // exact in f32

    v2f bones; bones.x = 1.0f; bones.y = 1.0f;           // B = all ones (4x16)
    v8f acc = {};                                        // C/D accumulator

    for (int kk = 0; kk < KS; kk += 4) {
        v2f afrag;
        #pragma unroll
        for (int u = 0; u < 2; ++u) {
            const int k = kk + kb + u;
            const float kf = (float)k;
            const float t  = (kf - ey) * invIy;          // grid_y(k) == k
            const float gx = fmaf(Ix, t, ex);            // world x -> indexes D
            const float gz = fmaf(Iz, t, ez);            // world z -> indexes H
            const float iyf = kf * cyc;                  // exact, always in [0,126.01]
            const int   y0  = (int)iyf;
            const float wy1 = iyf - (float)y0;
            const float wy0 = 1.0f - wy1;
            const float ixf = (gz + 64.0f) * cyc;        // H index
            const float izf = (gx + 64.0f) * cyc;        // D index
            const float s = tri_sample(vol, ixf, y0, wy0, wy1, izf) * dx;
            if (u == 0) afrag.x = s; else afrag.y = s;
        }
        // D = A x ones + C : accumulates per-ray K-sums in f32
        acc = __builtin_amdgcn_wmma_f32_16x16x4_f32(
                  false, afrag, false, bones, (short)0, acc, false, false);
    }

    // D layout (32-bit C/D 16x16): lane<16 holds rows 0-7 (acc[v] = D[v][lane]),
    // lane>=16 holds rows 8-15 (acc[v] = D[v+8][lane-16]). Columns identical.
    int sel = ((lane < 16) ? lane : (lane - 16)) & 7;
    float val = acc[0];
    #pragma unroll
    for (int v = 1; v < 8; ++v) val = (sel == v) ? acc[v] : val;

    const bool storer = (lane < 8) || (lane >= 16 && lane < 24);
    const int  jj = j0 + ((lane < 16) ? lane : (lane - 8));
    if (storer && jj < RES) {
        out[(b * RES + i) * RES + jj] = val;
    }
}

extern "C" void kernel_launch(void* const* d_in, const int* in_sizes, int n_in,
                              void* d_out, int out_size, void* d_ws, size_t ws_size,
                              hipStream_t stream) {
    const float* vol     = (const float*)d_in[0];   // [1,1,128,128,128] f32
    const float* poses   = (const float*)d_in[1];   // [4,3] f32
    const float* spacing = (const float*)d_in[2];   // [3] f32
    float* out = (float*)d_out;                     // [4,1,180,180] f32

    const int nWaves = NBATCH * RES * JTILES;       // 8640
    const int blocks = nWaves / 8;                  // 1080 (256 thr = 8 waves/blk)
    projection_kernel<<<blocks, 256, 0, stream>>>(vol, poses, spacing, out);
}